// EfficientMultiLatentAttention_33191507263685
// MI455X (gfx1250) — compile-verified
//
#include <hip/hip_runtime.h>
#include <hip/hip_bf16.h>
#include <math.h>
#include <stdint.h>

// ---------------------------------------------------------------------------
// MLA decode step for MI455X (gfx1250, wave32).
//   B=8, D=1024, LOW=512, S_PAST=8191, S=8192
// Memory-bound (~2.6 GB HBM traffic => ~115us floor @ 23.3 TB/s).
//  * seven tiny projection GEMMs  -> v_wmma_f32_16x16x4_f32
//  * 939 MB cache append          -> b128 grid-stride copy + global_prefetch
//  * scores GEMV                  -> float4 (b128) loads, wave reduce
//  * attn = sum w*v               -> async global->LDS double buffering
//                                    (global_load_async_to_lds_b128 +
//                                     s_wait_asynccnt), 8-way S split with
//                                     deterministic partial reduce
// ---------------------------------------------------------------------------

typedef __attribute__((ext_vector_type(2))) float v2f;
typedef __attribute__((ext_vector_type(8))) float v8f;

#define BB    8
#define DD    1024
#define LOWW  512
#define SPAST 8191
#define SS    8192       // 2^13
#define ASPLIT 8         // S-partitions for attn-v kernel
#define CHNK  16         // v rows per async LDS tile

// ---------------------------------------------------------------------------
// WMMA f32 GEMM:  Y[BB, N] = X[BB, K] @ W[K, N]   (M padded 8 -> 16)
// One wave per 16-wide N tile. ISA 32-bit fragment layouts:
//   A (16x4):  lanes 0-15: M=lane, regs {K+0,K+1}; lanes 16-31: regs {K+2,K+3}
//   B (4x16):  lanes 0-15: N=lane, regs {K+0,K+1}; lanes 16-31: regs {K+2,K+3}
//   C/D:       VGPR j: lanes 0-15 -> M=j (rows 0-7 == all live rows)
// Branch-free A masking (rows >= BB contribute 0), no exec juggling.
// ---------------------------------------------------------------------------
__global__ void wmma_gemm_f32(const float* __restrict__ X, int ldx,
                              const float* __restrict__ W,
                              float* __restrict__ Y, int ldy,
                              int K, int N) {
  const int wave = (blockIdx.x * blockDim.x + threadIdx.x) >> 5;
  const int lane = threadIdx.x & 31;
  const int n0   = wave * 16;
  if (n0 >= N) return;
  const int   half  = lane >> 4;            // 0: lanes 0-15, 1: lanes 16-31
  const int   l     = lane & 15;
  const float amask = (l < BB) ? 1.0f : 0.0f;
  const float* Arow = X + (long)(l < BB ? l : 0) * ldx + 2 * half;
  const float* Bcol = W + (long)(2 * half) * N + n0 + l;

  v8f acc = {};
#pragma unroll 4
  for (int k = 0; k < K; k += 4) {
    v2f av = *(const v2f*)(Arow + k);       // b64 load (8B aligned)
    v2f a;  a.x = av.x * amask;  a.y = av.y * amask;
    v2f b;
    b.x = Bcol[(long)k * N];
    b.y = Bcol[(long)k * N + N];
    // (neg_a, A, neg_b, B, c_mod, C, reuse_a, reuse_b)
    acc = __builtin_amdgcn_wmma_f32_16x16x4_f32(false, a, false, b,
                                                (short)0, acc, false, false);
  }
  if (half == 0) {                          // rows 0..7 live in lanes 0-15
#pragma unroll
    for (int j = 0; j < BB; ++j) Y[(long)j * ldy + n0 + l] = acc[j];
  }
}

// ---------------------------------------------------------------------------
// Cache append:  dst[b, 0:SPAST, :] = cache[b]; dst[b, SPAST, :] = cur[b]
// Pure shifts (W4 and S are powers of two); b128 copy + stream prefetch.
// ---------------------------------------------------------------------------
__global__ void append_cache(const float4* __restrict__ cache,
                             const float4* __restrict__ cur,
                             float4* __restrict__ dst,
                             int shift /* log2(row width in float4) */) {
  const int  W4     = 1 << shift;
  const long total  = ((long)BB * SS) << shift;
  const long stride = (long)gridDim.x * blockDim.x;
  for (long i = (long)blockIdx.x * blockDim.x + threadIdx.x; i < total; i += stride) {
    const long row = i >> shift;            // 0 .. BB*SS-1
    const int  col = (int)(i & (W4 - 1));
    const int  s   = (int)(row & (SS - 1));
    const long b   = row >> 13;             // SS = 2^13
    float4 v;
    if (s != SPAST) {
      const float4* src = cache + (((b * SPAST + s) << shift) + col);
      __builtin_prefetch(src + (8 << shift), 0, 0);   // global_prefetch, stream-ahead
      v = *src;
    } else {
      v = cur[(b << shift) + col];
    }
    dst[i] = v;
  }
}

// ---------------------------------------------------------------------------
// scores[b,s] = scale * dot(q[b,0:2D], concat(up[b,s,0:D], kr[b,s,0:D]))
// One wave per (b,s); float4 loads; shfl wave reduce.
// ---------------------------------------------------------------------------
__global__ void scores_kernel(const float* __restrict__ q,   // [B, 2D]
                              const float* __restrict__ up,  // [B, S, 2D]
                              const float* __restrict__ kr,  // [B, S, D]
                              float* __restrict__ scores,    // [B, S]
                              float scale) {
  const int gwave = (blockIdx.x * blockDim.x + threadIdx.x) >> 5;
  const int lane  = threadIdx.x & 31;
  if (gwave >= BB * SS) return;
  const int b = gwave >> 13;                // /SS
  const int s = gwave & (SS - 1);

  const float4* q4  = (const float4*)(q  + (long)b * 2 * DD);
  const float4* ku4 = (const float4*)(up + ((long)b * SS + s) * 2 * DD);
  const float4* kr4 = (const float4*)(kr + ((long)b * SS + s) * DD);

  float sum = 0.0f;
#pragma unroll
  for (int i = lane; i < DD / 4; i += 32) { // 8 iterations
    const float4 qa = q4[i],           ka = ku4[i];
    const float4 qb = q4[DD / 4 + i],  kb = kr4[i];
    sum += qa.x * ka.x + qa.y * ka.y + qa.z * ka.z + qa.w * ka.w;
    sum += qb.x * kb.x + qb.y * kb.y + qb.z * kb.z + qb.w * kb.w;
  }
#pragma unroll
  for (int off = 16; off > 0; off >>= 1)
    sum += __shfl_down(sum, off, 32);
  if (lane == 0) scores[(long)b * SS + s] = sum * scale;
}

// ---------------------------------------------------------------------------
// In-place softmax over S per batch. One 1024-thread block per batch.
// ---------------------------------------------------------------------------
__global__ void softmax_kernel(float* __restrict__ scores) {
  const int b = blockIdx.x;
  float* sc = scores + (long)b * SS;
  __shared__ float redA[32];
  __shared__ float redB[32];
  const int tid = threadIdx.x;
  const int nw  = blockDim.x >> 5;

  float m = -INFINITY;
  for (int i = tid; i < SS; i += blockDim.x) m = fmaxf(m, sc[i]);
#pragma unroll
  for (int off = 16; off > 0; off >>= 1) m = fmaxf(m, __shfl_down(m, off, 32));
  if ((tid & 31) == 0) redA[tid >> 5] = m;
  __syncthreads();
  if (tid < 32) {
    float v = (tid < nw) ? redA[tid] : -INFINITY;
#pragma unroll
    for (int off = 16; off > 0; off >>= 1) v = fmaxf(v, __shfl_down(v, off, 32));
    if (tid == 0) redA[0] = v;
  }
  __syncthreads();
  m = redA[0];

  float sum = 0.0f;
  for (int i = tid; i < SS; i += blockDim.x) {
    const float e = __expf(sc[i] - m);
    sc[i] = e;
    sum += e;
  }
#pragma unroll
  for (int off = 16; off > 0; off >>= 1) sum += __shfl_down(sum, off, 32);
  if ((tid & 31) == 0) redB[tid >> 5] = sum;
  __syncthreads();
  if (tid < 32) {
    float v = (tid < nw) ? redB[tid] : 0.0f;
#pragma unroll
    for (int off = 16; off > 0; off >>= 1) v += __shfl_down(v, off, 32);
    if (tid == 0) redB[0] = v;
  }
  __syncthreads();
  const float inv = 1.0f / redB[0];
  for (int i = tid; i < SS; i += blockDim.x) sc[i] *= inv;
}

// ---------------------------------------------------------------------------
// partial[split,b,d] = sum_{s in split} w[b,s] * v[b,s,d],  v = up[b,s,D:2D]
// v tiles (CHNK rows x 256 floats = 16 KB) staged via async global->LDS
// (global_load_async_to_lds_b128), double-buffered; s_wait_asynccnt 4 keeps
// the newest chunk in flight (async loads retire in order), then a barrier
// publishes the older chunk to all 8 waves.
// ---------------------------------------------------------------------------
__global__ void attnv_kernel(const float* __restrict__ w,    // [B,S] softmaxed
                             const float* __restrict__ up,   // [B,S,2D]
                             float* __restrict__ partial) {  // [ASPLIT,B,D]
  const int b     = blockIdx.y;
  const int split = blockIdx.z;
  const int d0    = blockIdx.x * 256;
  const int tid   = threadIdx.x;

  __shared__ float buf[2][CHNK * 256];      // 32 KB double buffer

  const int s_begin = split * (SS / ASPLIT);
  const int nch     = (SS / ASPLIT) / CHNK; // 64 chunks of 16 rows

  auto issue = [&](int c) {
    const int s0 = s_begin + c * CHNK;
    const float* gbase = up + ((long)b * SS + s0) * (2 * DD) + DD + d0;
    float* lbase = &buf[c & 1][0];
#pragma unroll
    for (int i = 0; i < 4; ++i) {
      const int flat = tid + i * 256;       // float4 slot 0..1023
      const int r    = flat >> 6;           // row in chunk
      const int col  = flat & 63;           // float4 column
      const float* g = gbase + (long)r * (2 * DD) + col * 4;
      const unsigned lds = (unsigned)(uintptr_t)(lbase + flat * 4);
      asm volatile("global_load_async_to_lds_b128 %0, %1, off"
                   :: "v"(lds), "v"(g) : "memory");
    }
  };

  issue(0);
  float acc = 0.0f;
  const float* wb = w + (long)b * SS + s_begin;
  for (int c = 0; c < nch; ++c) {
    if (c + 1 < nch) {
      issue(c + 1);
      asm volatile("s_wait_asynccnt 0x4" ::: "memory"); // older chunk complete
    } else {
      asm volatile("s_wait_asynccnt 0x0" ::: "memory");
    }
    __syncthreads();                        // publish LDS tile across waves
    const float* lb = &buf[c & 1][0];
    const float* wc = wb + c * CHNK;
#pragma unroll
    for (int r = 0; r < CHNK; ++r)
      acc += wc[r] * lb[r * 256 + tid];
    __syncthreads();                        // consumption done before reuse
  }
  partial[((long)split * BB + b) * DD + d0 + tid] = acc;
}

__global__ void attnv_reduce(const float* __restrict__ partial, // [ASPLIT,B,D]
                             float* __restrict__ attn) {        // [B,D]
  const int idx = blockIdx.x * blockDim.x + threadIdx.x;
  if (idx >= BB * DD) return;
  float s = 0.0f;
#pragma unroll
  for (int p = 0; p < ASPLIT; ++p) s += partial[(long)p * BB * DD + idx];
  attn[idx] = s;
}

// ---------------------------------------------------------------------------
// Launch
// ---------------------------------------------------------------------------
extern "C" void kernel_launch(void* const* d_in, const int* in_sizes, int n_in,
                              void* d_out, int out_size, void* d_ws, size_t ws_size,
                              hipStream_t stream) {
  const float* x     = (const float*)d_in[0];   // [B,1,D]
  const float* ckv_c = (const float*)d_in[1];   // [B,SPAST,LOW]
  const float* kr_c  = (const float*)d_in[2];   // [B,SPAST,D]
  const float* up_c  = (const float*)d_in[3];   // [B,SPAST,2D]
  const float* Wdq   = (const float*)d_in[4];
  const float* Wuq   = (const float*)d_in[5];
  const float* Wqr   = (const float*)d_in[6];
  const float* Wdkv  = (const float*)d_in[7];
  const float* Wkr   = (const float*)d_in[8];
  const float* Wfup  = (const float*)d_in[9];   // [LOW,2D]
  const float* Wo    = (const float*)d_in[10];

  float* out   = (float*)d_out;                         // [B,D]
  float* ckv_o = out + (long)BB * DD;                   // [B,S,LOW]
  float* kr_o  = ckv_o + (long)BB * SS * LOWW;          // [B,S,D]
  float* up_o  = kr_o + (long)BB * SS * DD;             // [B,S,2D]

  // Workspace layout (floats)
  float* ws      = (float*)d_ws;
  float* Cq      = ws;                                  // [B,D]
  float* q       = Cq + BB * DD;                        // [B,2D]
  float* cur_ckv = q + BB * 2 * DD;                     // [B,LOW]
  float* cur_kr  = cur_ckv + BB * LOWW;                 // [B,D]
  float* cur_up  = cur_kr + BB * DD;                    // [B,2D]
  float* scores  = cur_up + BB * 2 * DD;                // [B,S]
  float* attn    = scores + (long)BB * SS;              // [B,D]
  float* partial = attn + BB * DD;                      // [ASPLIT,B,D]

  // --- Projections (WMMA f32 16x16x4) --------------------------------------
  const int GB = 128;                                   // 4 waves / block
  auto tiles = [](int N) { return (N / 16 + 3) / 4; };

  hipLaunchKernelGGL(wmma_gemm_f32, dim3(tiles(DD)), dim3(GB), 0, stream,
                     x, DD, Wdq, Cq, DD, DD, DD);                    // Cq = x@Wdq
  hipLaunchKernelGGL(wmma_gemm_f32, dim3(tiles(DD)), dim3(GB), 0, stream,
                     x, DD, Wkr, cur_kr, DD, DD, DD);                // cur_kr = x@Wkr
  hipLaunchKernelGGL(wmma_gemm_f32, dim3(tiles(LOWW)), dim3(GB), 0, stream,
                     x, DD, Wdkv, cur_ckv, LOWW, DD, LOWW);          // cur_ckv = x@Wdkv
  hipLaunchKernelGGL(wmma_gemm_f32, dim3(tiles(DD)), dim3(GB), 0, stream,
                     Cq, DD, Wuq, q, 2 * DD, DD, DD);                // Qc -> q[:,0:D]
  hipLaunchKernelGGL(wmma_gemm_f32, dim3(tiles(DD)), dim3(GB), 0, stream,
                     Cq, DD, Wqr, q + DD, 2 * DD, DD, DD);           // Qr -> q[:,D:2D]
  hipLaunchKernelGGL(wmma_gemm_f32, dim3(tiles(2 * DD)), dim3(GB), 0, stream,
                     cur_ckv, LOWW, Wfup, cur_up, 2 * DD, LOWW, 2 * DD); // cur_up

  // --- Cache append (bulk b128 copies) -------------------------------------
  hipLaunchKernelGGL(append_cache, dim3(4096), dim3(256), 0, stream,
                     (const float4*)ckv_c, (const float4*)cur_ckv,
                     (float4*)ckv_o, 7);                             // 512/4 = 2^7
  hipLaunchKernelGGL(append_cache, dim3(8192), dim3(256), 0, stream,
                     (const float4*)kr_c, (const float4*)cur_kr,
                     (float4*)kr_o, 8);                              // 1024/4 = 2^8
  hipLaunchKernelGGL(append_cache, dim3(16384), dim3(256), 0, stream,
                     (const float4*)up_c, (const float4*)cur_up,
                     (float4*)up_o, 9);                              // 2048/4 = 2^9

  // --- Attention -----------------------------------------------------------
  const float scale = 1.0f / sqrtf(2.0f * (float)DD);
  hipLaunchKernelGGL(scores_kernel, dim3((BB * SS) / 8), dim3(256), 0, stream,
                     q, up_o, kr_o, scores, scale);
  hipLaunchKernelGGL(softmax_kernel, dim3(BB), dim3(1024), 0, stream, scores);
  hipLaunchKernelGGL(attnv_kernel, dim3(DD / 256, BB, ASPLIT), dim3(256), 0, stream,
                     scores, up_o, partial);
  hipLaunchKernelGGL(attnv_reduce, dim3((BB * DD) / 256), dim3(256), 0, stream,
                     partial, attn);
  hipLaunchKernelGGL(wmma_gemm_f32, dim3(tiles(DD)), dim3(GB), 0, stream,
                     attn, DD, Wo, out, DD, DD, DD);                 // out = attn@Wo
}